// DistanceLossMultipleMargins_25890062860768
// MI455X (gfx1250) — compile-verified
//
#include <hip/hip_runtime.h>

// CDNA5 / gfx1250: wave32, WMMA 16x16x4 f32 used as a row-sum engine.
typedef __attribute__((ext_vector_type(2))) float v2f;
typedef __attribute__((ext_vector_type(8))) float v8f;

#define DIM       256
#define CHUNK     64
#define PAD       4
#define RSTRIDE   (CHUNK + PAD)          // 68 floats -> bank-conflict-free rows
#define TILE_ROWS 16
#define NWAVES    8
#define ROWS_PER_BLOCK (TILE_ROWS * NWAVES)  // 128 rows per 256-thread block
#define NCLS      8

__global__ void zero_out_kernel(float* out) {
    if (threadIdx.x == 0) out[0] = 0.0f;
}

__global__ __launch_bounds__(256, 2)
void triplet_margin_loss_kernel(const float* __restrict__ A,
                                const float* __restrict__ P,
                                const float* __restrict__ Ng,
                                const int*   __restrict__ AL,
                                const int*   __restrict__ NL,
                                const float* __restrict__ MM,
                                float* __restrict__ out,
                                int batch)
{
    __shared__ float tileLds[NWAVES][TILE_ROWS * RSTRIDE];

    const int lane    = threadIdx.x & 31;
    const int wave    = threadIdx.x >> 5;
    const int rowBase = blockIdx.x * ROWS_PER_BLOCK + wave * TILE_ROWS;
    if (rowBase >= batch) return;        // wave-uniform: EXEC stays all-ones for WMMA

    float* lds = tileLds[wave];

    float mse = 0.0f;                    // per-lane partial of sum((A-P)^2)
    v8f acc = {0.f, 0.f, 0.f, 0.f, 0.f, 0.f, 0.f, 0.f};
    v2f ones; ones[0] = 1.0f; ones[1] = 1.0f;   // B = 4x16 ones -> D cols = row sums

    // A-fragment mapping for V_WMMA_F32_16X16X4_F32:
    // lanes 0-15 hold row M=lane, K = {0,1}; lanes 16-31 hold row M=lane-16, K = {2,3}
    const int arow = lane & 15;
    const int kk   = (lane >> 4) << 1;

    for (int c0 = 0; c0 < DIM; c0 += CHUNK) {
        // ---- Phase A: coalesced 128-bit streaming loads; MSE + diff^2 -> LDS ----
#pragma unroll
        for (int i = 0; i < 8; ++i) {
            const int linear = i * 128 + lane * 4;        // 0..1023 over 16x64 tile
            const int r  = linear >> 6;                   // tile row 0..15
            const int cc = linear & 63;                   // tile col 0..63
            const size_t off = (size_t)(rowBase + r) * DIM + (size_t)(c0 + cc);
            const float4 av = *reinterpret_cast<const float4*>(A  + off);
            const float4 pv = *reinterpret_cast<const float4*>(P  + off);
            const float4 nv = *reinterpret_cast<const float4*>(Ng + off);

            const float dpx = av.x - pv.x, dpy = av.y - pv.y;
            const float dpz = av.z - pv.z, dpw = av.w - pv.w;
            mse += dpx * dpx + dpy * dpy + dpz * dpz + dpw * dpw;

            float4 dq;
            const float dnx = av.x - nv.x, dny = av.y - nv.y;
            const float dnz = av.z - nv.z, dnw = av.w - nv.w;
            dq.x = dnx * dnx; dq.y = dny * dny; dq.z = dnz * dnz; dq.w = dnw * dnw;
            *reinterpret_cast<float4*>(&lds[r * RSTRIDE + cc]) = dq;
        }

        // ---- Phase B: WMMA row-sum; D += A(16x4 diff^2) x B(4x16 ones) ----
        // Same-wave LDS ops are hardware in-order; no barrier needed.
#pragma unroll
        for (int kc = 0; kc < CHUNK / 4; ++kc) {
            const v2f afrag =
                *reinterpret_cast<const v2f*>(&lds[arow * RSTRIDE + kc * 4 + kk]);
            acc = __builtin_amdgcn_wmma_f32_16x16x4_f32(
                false, afrag, false, ones, (short)0, acc, false, false);
        }
    }

    // ---- Extract per-row sums from the 16x16 accumulator ----
    // C/D layout: VGPR j, lanes 0-15 -> M=j ; lanes 16-31 -> M=j+8. All N equal.
    const bool lo = (lane < 8);                     // rows 0..7  via acc[lane]
    const bool hi = (lane >= 16) && (lane < 24);    // rows 8..15 via acc[lane-16]
    const int  sel = lane & 7;
    const float rowsum = acc[sel];

    int rowIdx = rowBase + (lo ? lane : (hi ? (lane - 8) : 0));
    if (rowIdx >= batch) rowIdx = batch - 1;        // safety clamp (full-EXEC loads)

    const float dist = sqrtf(rowsum);
    const int   al   = AL[rowIdx];
    const int   nl   = NL[rowIdx];
    const float mg   = MM[al * NCLS + nl];
    const float res  = mg - dist;
    float negc = (lo || hi) ? res * res : 0.0f;

    // ---- wave32 butterfly reduction, one atomic per wave ----
#pragma unroll
    for (int offs = 16; offs > 0; offs >>= 1) {
        negc += __shfl_xor(negc, offs, 32);
        mse  += __shfl_xor(mse,  offs, 32);
    }
    if (lane == 0) {
        const float invB  = 1.0f / (float)batch;
        const float invBD = invB / (float)DIM;
        atomicAdd(out, mse * invBD + negc * invB);
    }
}

extern "C" void kernel_launch(void* const* d_in, const int* in_sizes, int n_in,
                              void* d_out, int out_size, void* d_ws, size_t ws_size,
                              hipStream_t stream) {
    const float* A  = (const float*)d_in[0];   // anchor_output   [B, 256]
    const float* P  = (const float*)d_in[1];   // positive_output [B, 256]
    const float* Ng = (const float*)d_in[2];   // negative_output [B, 256]
    const int*   AL = (const int*)  d_in[3];   // anchor_label    [B] (int32)
    const int*   NL = (const int*)  d_in[4];   // negative_label  [B] (int32)
    const float* MM = (const float*)d_in[5];   // margin_matrix   [8, 8]
    float* out = (float*)d_out;

    const int batch  = in_sizes[0] / DIM;
    const int blocks = (batch + ROWS_PER_BLOCK - 1) / ROWS_PER_BLOCK;

    zero_out_kernel<<<1, 1, 0, stream>>>(out);
    triplet_margin_loss_kernel<<<blocks, 256, 0, stream>>>(A, P, Ng, AL, NL, MM,
                                                           out, batch);
}